// BottomUp_57131654972209
// MI455X (gfx1250) — compile-verified
//
#include <hip/hip_runtime.h>
#include <stdint.h>

#define BLK 64
#define NL 60
#define NC 48

// Low 32 bits of a generic (flat) address of a __shared__ object are the LDS
// byte offset (ISA 10.2).
__device__ __forceinline__ uint32_t lds_off_of(const void* p) {
  return (uint32_t)(uintptr_t)p;
}

// CDNA5 async global->LDS copy, GV addressing (64-bit VGPR address, SADDR=off).
// Tracked with ASYNCcnt. (CDNA5 ISA)
__device__ __forceinline__ void async_ld_b32(uint32_t lds_off, const float* g) {
  asm volatile("global_load_async_to_lds_b32 %0, %1, off"
               :: "v"(lds_off), "v"(g)
               : "memory");
}

__device__ __forceinline__ void wait_async_le4() {
  asm volatile("s_wait_asynccnt 0x4" ::: "memory");
}
__device__ __forceinline__ void wait_async_0() {
  asm volatile("s_wait_asynccnt 0x0" ::: "memory");
}

__global__ __launch_bounds__(BLK) void adding_doubling_kernel(
    const float* __restrict__ a, const float* __restrict__ r,
    const float* __restrict__ t, const float* __restrict__ s,
    float* __restrict__ flux_up, float* __restrict__ flux_down,
    float* __restrict__ absorbed) {
  // Per-thread scan state parked in LDS (each thread touches only its own slot
  // -> no barriers; avoids scratch spills for the 59-deep recurrences).
  __shared__ float rsl[NL - 1][BLK];   // rs_below[l] per thread
  __shared__ float aup[NL - 1][BLK];   // absorbed_up[l] per thread
  __shared__ float stage[2][4][BLK];   // double-buffered async staging: a,r,t,s

  const int tid = threadIdx.x;
  const int j = blockIdx.x * BLK + tid;          // column id in [0, E*C)
  const int e = j / NC;
  const int c = j % NC;

  const size_t colbase = (size_t)e * (NL * NC) + c;   // input (E,L,C) stride C per layer
  const float* pa = a + colbase;
  const float* pr = r + colbase;
  const float* pt = t + colbase;
  const float* ps = s + colbase;

  const size_t obase = (size_t)e * ((NL - 1) * NC) + c; // output (E,L-1,C)
  float* pfu = flux_up + obase;
  float* pfd = flux_down + obase;
  float* pab = absorbed + obase;

  // ---- layer L-1 seed of the reverse reflection scan ----
  const float r59 = pr[(NL - 1) * NC];
  const float t59 = pt[(NL - 1) * NC];
  float prev_s = ps[(NL - 1) * NC];
  float lr = r59;                               // carry = r[:, -1, :]
  {
    const float dd = 1.0f - lr * r59;           // d[L-1] (unused downstream)
    lr = (r59 + lr * t59 * t59) / dd;           // rs_body[L-1] -> rs_below[L-2]
  }
  float carry_up = 0.0f;

  // Pre-issue async staging for layer 58 (a, r, t, s).
  {
    const int l = NL - 2;
    const int p = l & 1;
    async_ld_b32(lds_off_of(&stage[p][0][tid]), pa + l * NC);
    async_ld_b32(lds_off_of(&stage[p][1][tid]), pr + l * NC);
    async_ld_b32(lds_off_of(&stage[p][2][tid]), pt + l * NC);
    async_ld_b32(lds_off_of(&stage[p][3][tid]), ps + l * NC);
  }

  // ---- backward pass: surface reflection + upward flux, fused ----
  for (int l = NL - 2; l >= 0; --l) {
    if (l > 0) {
      const int q = (l - 1) & 1;                // stage next layer while this one waits
      async_ld_b32(lds_off_of(&stage[q][0][tid]), pa + (l - 1) * NC);
      async_ld_b32(lds_off_of(&stage[q][1][tid]), pr + (l - 1) * NC);
      async_ld_b32(lds_off_of(&stage[q][2][tid]), pt + (l - 1) * NC);
      async_ld_b32(lds_off_of(&stage[q][3][tid]), ps + (l - 1) * NC);
      wait_async_le4();                         // layer l's 4 loads (in-order) done
    } else {
      wait_async_0();
    }
    const int p = l & 1;
    const float al = stage[p][0][tid];
    const float rl = stage[p][1][tid];
    const float tl = stage[p][2][tid];
    const float sl = stage[p][3][tid];

    const float tmp = lr * rl;                  // rs_below[l] * r[l]
    const float inv1p = 1.0f / (1.0f + tmp);
    const float ddl = 1.0f - tmp;               // d[l] = 1 - rs_below[l]*r[l]

    const float s_multi_up = prev_s * (1.0f + tmp * inv1p);
    const float s_multi_down_up = sl * lr * inv1p;
    const float fu = carry_up + s_multi_up + s_multi_down_up;
    __builtin_nontemporal_store(fu, &pfu[l * NC]);   // flux_up[l], write-once -> NT
    aup[l][tid] = al * fu;                           // absorbed_up[l]
    carry_up = fu * tl / ddl;

    rsl[l][tid] = lr;                           // save rs_below[l] for forward pass
    lr = (rl + lr * tl * tl) / ddl;             // rs_body[l]
    prev_s = sl;
  }

  // ---- forward pass: downward flux + absorbed ----
  // Re-reads are WGP$/L2-warm; mark them non-temporal (last use) so the lines
  // retire instead of aging out other blocks' first-touch data.
  float carry_down = 0.0f;
  float s_cur = __builtin_nontemporal_load(&ps[0]);
  for (int l = 0; l < NL - 1; ++l) {
    const float al = __builtin_nontemporal_load(&pa[l * NC]);
    const float rl = __builtin_nontemporal_load(&pr[l * NC]);
    const float tl = __builtin_nontemporal_load(&pt[l * NC]);
    const float s_next = __builtin_nontemporal_load(&ps[(l + 1) * NC]);
    const float lrv = rsl[l][tid];

    const float tmp = lrv * rl;
    const float inv1p = 1.0f / (1.0f + tmp);
    const float inv1m = 1.0f / (1.0f - tmp);

    const float t_multi = tl * inv1p;
    const float s_multi_down = s_cur * (1.0f + tmp * inv1m);
    const float s_multi_up_down = s_next * rl * inv1m;

    const float fd = carry_down + s_multi_down + s_multi_up_down;
    __builtin_nontemporal_store(fd, &pfd[l * NC]);   // flux_down[l]

    const float a_multi = al * (1.0f + tl * lrv * inv1p);
    const float ab = a_multi * fd + aup[l][tid];
    __builtin_nontemporal_store(ab, &pab[l * NC]);   // absorbed[l]

    carry_down = fd * t_multi;
    s_cur = s_next;
  }
}

extern "C" void kernel_launch(void* const* d_in, const int* in_sizes, int n_in,
                              void* d_out, int out_size, void* d_ws, size_t ws_size,
                              hipStream_t stream) {
  (void)n_in; (void)d_ws; (void)ws_size; (void)out_size;
  const float* a = (const float*)d_in[0];
  const float* r = (const float*)d_in[1];
  const float* t = (const float*)d_in[2];
  const float* s = (const float*)d_in[3];

  const int total = in_sizes[0];        // E*L*C
  const int ncol = total / NL;          // E*C (divisible by BLK: 8192*48/64)
  const int E = ncol / NC;
  const size_t osz = (size_t)E * (NL - 1) * NC;

  float* fu = (float*)d_out;
  float* fd = fu + osz;
  float* ab = fd + osz;

  dim3 grid(ncol / BLK), block(BLK);
  hipLaunchKernelGGL(adding_doubling_kernel, grid, block, 0, stream,
                     a, r, t, s, fu, fd, ab);
}